// RecurrentAutoencoder_75213467288202
// MI455X (gfx1250) — compile-verified
//
#include <hip/hip_runtime.h>
#include <math.h>

// ---------------------------------------------------------------------------
// LSTM-VAE forward for MI455X (gfx1250, wave32).
//  * Input projections hoisted out of the recurrence -> big f32 WMMA GEMMs
//    (V_WMMA_F32_16X16X4_F32: exact f32 precision, matches the reference).
//  * K==1 input projection (enc layer 0) is an outer product -> scalar kernel.
//  * Recurrence: persistent kernel, Whh cached in LDS, h double-buffered in
//    LDS, c entirely in registers; one __syncthreads per timestep.
// Workspace requirement: ~118 MB (layout in kernel_launch).
// ---------------------------------------------------------------------------

typedef float v2f __attribute__((ext_vector_type(2)));
typedef float v8f __attribute__((ext_vector_type(8)));

#define SEQ   512
#define BATCH 64

__device__ __forceinline__ float sigm(float x) { return 1.f / (1.f + expf(-x)); }

// ---------------------------------------------------------------------------
// C[M,N] = A[M,K] (row-major, row stride lda) @ W[N,K]^T + ba[N] + bb[N]
// REQUIRES: K % 4 == 0, lda even (8B-aligned v2f loads), M,N multiples of 16.
// One wave per 16x16 output tile; K consumed 4 per WMMA.
// A-operand (32-bit A 16x4): lanes 0-15: M=lane, K={k0,k0+1};
//                            lanes 16-31: M=lane-16, K={k0+2,k0+3}.
// B-operand mirrors over columns. D: VGPR v -> row v (+8 for lanes>=16),
// col = lane&15.
// ---------------------------------------------------------------------------
__global__ void wmma_gemm_bias(const float* __restrict__ A, int lda,
                               const float* __restrict__ W,
                               const float* __restrict__ ba,
                               const float* __restrict__ bb,
                               float* __restrict__ C, int M, int N, int K)
{
    const int lane = threadIdx.x & 31;
    const int wave = threadIdx.x >> 5;
    const int ntn  = N >> 4;
    const int tile = blockIdx.x * (blockDim.x >> 5) + wave;
    if (tile >= (M >> 4) * ntn) return;                 // whole-wave exit only
    const int m0 = (tile / ntn) << 4;
    const int n0 = (tile % ntn) << 4;
    const int r  = lane & 15;
    const int kh = (lane >> 4) << 1;

    const float* arow = A + (m0 + r) * lda + kh;
    const float* wrow = W + (n0 + r) * K + kh;

    v8f acc = {};
#pragma unroll 4
    for (int k0 = 0; k0 < K; k0 += 4) {
        const v2f av = *(const v2f*)(arow + k0);
        const v2f bv = *(const v2f*)(wrow + k0);
        acc = __builtin_amdgcn_wmma_f32_16x16x4_f32(false, av, false, bv,
                                                    (short)0, acc, false, false);
    }
    const float bias  = ba[n0 + r] + bb[n0 + r];
    const int   mbase = m0 + ((lane >> 4) << 3);
#pragma unroll
    for (int v = 0; v < 8; ++v)
        C[(mbase + v) * N + (n0 + r)] = acc[v] + bias;
}

// ---------------------------------------------------------------------------
// K==1 input projection (outer product): C[m,n] = A[m]*W[n] + ba[n] + bb[n]
// ---------------------------------------------------------------------------
__global__ void gemm_k1(const float* __restrict__ A, const float* __restrict__ W,
                        const float* __restrict__ ba, const float* __restrict__ bb,
                        float* __restrict__ C, int M, int N)
{
    const int i = blockIdx.x * blockDim.x + threadIdx.x;
    if (i >= M * N) return;
    const int m = i / N, n = i - m * N;
    C[i] = fmaf(A[m], W[n], ba[n] + bb[n]);
}

// ---------------------------------------------------------------------------
// Persistent LSTM recurrence. grid.x = #directions (1 or 2), block = 256 (8
// waves). Each wave owns NG/8 (batch-tile x hidden-tile) groups and all four
// gates for them: c never leaves registers. h double-buffered in LDS; Whh
// staged in LDS once (reused T times). gx = precomputed x_t@Wih^T + bih+bhh,
// addressed gx[b*gx_sb + t*gx_st + col] (gx_st==0 => time-constant input).
// ---------------------------------------------------------------------------
template <int H>
__global__ void lstm_rec(const float* __restrict__ gx0, const float* __restrict__ gx1,
                         int gx_sb, int gx_st,
                         const float* __restrict__ whh0, const float* __restrict__ whh1,
                         float* __restrict__ y, int y_stride, int ycol0, int ycol1,
                         float* __restrict__ hT, int hT_stride, int hTcol0, int hTcol1,
                         int T, int rev0, int rev1)
{
    constexpr int NTN = H / 16;   // hidden tiles per gate
    constexpr int NG  = 4 * NTN;  // (batch 64 / 16) * NTN tile-groups
    constexpr int GPW = NG / 8;   // groups per wave

    __shared__ float hbuf[2][BATCH * H];
    __shared__ float wlds[4 * H * H];

    const int dir = blockIdx.x;
    const float* gx  = dir ? gx1 : gx0;
    const float* whh = dir ? whh1 : whh0;
    const int ycol   = dir ? ycol1 : ycol0;
    const int hTcol  = dir ? hTcol1 : hTcol0;
    const int rev    = dir ? rev1 : rev0;

    const int tid  = threadIdx.x;
    const int lane = tid & 31;
    const int wave = tid >> 5;
    const int r    = lane & 15;
    const int kh   = (lane >> 4) << 1;
    const int voff = (lane >> 4) << 3;

    for (int i = tid; i < BATCH * H; i += 256) hbuf[0][i] = 0.f;
    for (int i = tid; i < 4 * H * H; i += 256) wlds[i] = whh[i];
    __syncthreads();

    v8f cst[GPW] = {};

    for (int s = 0; s < T; ++s) {
        const int t = rev ? (T - 1 - s) : s;
        const float* hin  = hbuf[s & 1];
        float*       hout = hbuf[(s + 1) & 1];

#pragma unroll
        for (int gidx = 0; gidx < GPW; ++gidx) {
            const int grp = wave + (gidx << 3);
            const int m0  = (grp / NTN) << 4;   // batch tile base
            const int nh0 = (grp % NTN) << 4;   // hidden tile base

            const float* hrow = hin + (m0 + r) * H + kh;
            const float* wrow = wlds + (nh0 + r) * H + kh;

            v8f acc0 = {}, acc1 = {}, acc2 = {}, acc3 = {};
#pragma unroll
            for (int k0 = 0; k0 < H; k0 += 4) {
                const v2f a  = *(const v2f*)(hrow + k0);
                const v2f b0 = *(const v2f*)(wrow + k0);
                const v2f b1 = *(const v2f*)(wrow + H * H + k0);
                const v2f b2 = *(const v2f*)(wrow + 2 * H * H + k0);
                const v2f b3 = *(const v2f*)(wrow + 3 * H * H + k0);
                acc0 = __builtin_amdgcn_wmma_f32_16x16x4_f32(false, a, false, b0, (short)0, acc0, false, false);
                acc1 = __builtin_amdgcn_wmma_f32_16x16x4_f32(false, a, false, b1, (short)0, acc1, false, false);
                acc2 = __builtin_amdgcn_wmma_f32_16x16x4_f32(false, a, false, b2, (short)0, acc2, false, false);
                acc3 = __builtin_amdgcn_wmma_f32_16x16x4_f32(false, a, false, b3, (short)0, acc3, false, false);
            }

            const int ncol = nh0 + r;  // hidden index
#pragma unroll
            for (int v = 0; v < 8; ++v) {
                const int b = m0 + voff + v;  // batch index
                const float* grow = gx + b * gx_sb + t * gx_st;
                const float iv = sigm (acc0[v] + grow[ncol]);
                const float fv = sigm (acc1[v] + grow[H + ncol]);
                const float gv = tanhf(acc2[v] + grow[2 * H + ncol]);
                const float ov = sigm (acc3[v] + grow[3 * H + ncol]);
                float c = fv * cst[gidx][v] + iv * gv;
                cst[gidx][v] = c;
                const float h = ov * tanhf(c);
                hout[b * H + ncol] = h;
                y[(b * T + t) * y_stride + ycol + ncol] = h;
            }
        }
        __syncthreads();
    }

    if (hT) {
        for (int i = tid; i < BATCH * H; i += 256) {
            const int b = i / H, hh = i % H;
            hT[b * hT_stride + hTcol + hh] = hbuf[T & 1][i];
        }
    }
}

// ---------------------------------------------------------------------------
// H==1 unidirectional tail (dec_rnn3): 4H=4 gates, scalar per batch lane.
// ---------------------------------------------------------------------------
__global__ void lstm_h1(const float* __restrict__ x, int xw,
                        const float* __restrict__ wih, const float* __restrict__ whh,
                        const float* __restrict__ bih, const float* __restrict__ bhh,
                        float* __restrict__ out, int T)
{
    __shared__ float w[4 * 128];
    __shared__ float bias[4];
    const int tid = threadIdx.x;
    for (int i = tid; i < 4 * xw; i += blockDim.x) w[i] = wih[i];
    if (tid < 4) bias[tid] = bih[tid] + bhh[tid];
    __syncthreads();
    if (tid >= BATCH) return;

    float h = 0.f, c = 0.f;
    const float wh0 = whh[0], wh1 = whh[1], wh2 = whh[2], wh3 = whh[3];
    for (int t = 0; t < T; ++t) {
        const float* xr = x + (tid * T + t) * xw;
        float s0 = bias[0] + wh0 * h, s1 = bias[1] + wh1 * h;
        float s2 = bias[2] + wh2 * h, s3 = bias[3] + wh3 * h;
        for (int k = 0; k < xw; ++k) {
            const float xv = xr[k];
            s0 += w[k] * xv;          s1 += w[xw + k] * xv;
            s2 += w[2 * xw + k] * xv; s3 += w[3 * xw + k] * xv;
        }
        c = sigm(s1) * c + sigm(s0) * tanhf(s2);
        h = sigm(s3) * tanhf(c);
        out[tid * T + t] = h;
    }
}

__global__ void reparam(const float* __restrict__ mu, const float* __restrict__ lv,
                        const float* __restrict__ eps, float* __restrict__ z, int n)
{
    const int i = blockIdx.x * blockDim.x + threadIdx.x;
    if (i < n) z[i] = mu[i] + expf(0.5f * lv[i]) * eps[i];
}

// ---------------------------------------------------------------------------
extern "C" void kernel_launch(void* const* d_in, const int* in_sizes, int n_in,
                              void* d_out, int out_size, void* d_ws, size_t ws_size,
                              hipStream_t stream)
{
    (void)in_sizes; (void)out_size; (void)ws_size;

    // ---- resolve parameter leaves; support both pytree flattenings --------
    const float* leaf[118];
    if (n_in >= 118) {
        for (int i = 0; i < 118; ++i) leaf[i] = (const float*)d_in[i];
    } else {
        leaf[0] = (const float*)d_in[0];
        leaf[1] = (const float*)d_in[1];
        int li = 2;
        auto fill = [&](const float* p, int nl, int D, int H, int in0, int inL) {
            for (int l = 0; l < nl; ++l) {
                const int isz = (l == 0) ? in0 : inL;
                for (int d = 0; d < D; ++d) {
                    leaf[li++] = p; p += 4 * H * isz;  // Wih
                    leaf[li++] = p; p += 4 * H * H;    // Whh
                    leaf[li++] = p; p += 4 * H;        // bih
                    leaf[li++] = p; p += 4 * H;        // bhh
                }
            }
        };
        fill((const float*)d_in[2], 4, 2, 32, 1, 64);     // enc_rnn1
        fill((const float*)d_in[3], 4, 2, 64, 64, 128);   // enc_mean
        fill((const float*)d_in[4], 4, 2, 64, 64, 128);   // enc_var
        fill((const float*)d_in[5], 1, 2, 32, 512, 512);  // dec_rnn1
        fill((const float*)d_in[6], 1, 2, 64, 64, 64);    // dec_rnn2
        fill((const float*)d_in[7], 1, 1, 1, 128, 128);   // dec_rnn3
    }
    const float* x   = leaf[0];
    const float* eps = leaf[1];
    const int ENC1 = 2, ENCM = 34, ENCV = 66, DEC1 = 98, DEC2 = 106, DEC3 = 114;
    auto P = [&](int i) { return leaf[i]; };

    // ---- workspace layout (floats) ---------------------------------------
    float* ws   = (float*)d_ws;
    float* bufA = ws;                            // 64*512*128
    float* bufB = bufA + BATCH * SEQ * 128;      // 64*512*128
    float* bufC = bufB + BATCH * SEQ * 128;      // 64*512*128
    float* gxa  = bufC + BATCH * SEQ * 128;      // 32768*256
    float* gxb  = gxa + BATCH * SEQ * 256;       // 32768*256
    float* zbuf = gxb + BATCH * SEQ * 256;       // 64*512
    float* gxc0 = zbuf + BATCH * 512;            // 64*128
    float* gxc1 = gxc0 + BATCH * 128;            // 64*128

    float* outD  = (float*)d_out;        // d        [64,512,1]
    float* outMu = outD + BATCH * SEQ;   // mu       [64,512]
    float* outLv = outMu + BATCH * 512;  // log_var  [64,512]

    auto gemm = [&](const float* A, int lda, const float* W, const float* ba,
                    const float* bb, float* C, int M, int N, int K) {
        const int tiles = (M / 16) * (N / 16);
        wmma_gemm_bias<<<(tiles + 7) / 8, 256, 0, stream>>>(A, lda, W, ba, bb, C, M, N, K);
    };

    // ---- enc_rnn1: 4-layer bidir, H=32 -----------------------------------
    {
        const float* in = x; int inw = 1;
        float* pp[2] = { bufA, bufB };
        for (int l = 0; l < 4; ++l) {
            const int b = ENC1 + l * 8;
            if (l == 0) {   // K==1: outer product
                const int n = BATCH * SEQ * 128;
                gemm_k1<<<(n + 255) / 256, 256, 0, stream>>>(in, P(b + 0), P(b + 2), P(b + 3), gxa, BATCH * SEQ, 128);
                gemm_k1<<<(n + 255) / 256, 256, 0, stream>>>(in, P(b + 4), P(b + 6), P(b + 7), gxb, BATCH * SEQ, 128);
            } else {
                gemm(in, inw, P(b + 0), P(b + 2), P(b + 3), gxa, BATCH * SEQ, 128, inw);
                gemm(in, inw, P(b + 4), P(b + 6), P(b + 7), gxb, BATCH * SEQ, 128, inw);
            }
            float* out = pp[l & 1];
            lstm_rec<32><<<2, 256, 0, stream>>>(gxa, gxb, SEQ * 128, 128,
                                                P(b + 1), P(b + 5),
                                                out, 64, 0, 32,
                                                nullptr, 0, 0, 0, SEQ, 0, 1);
            in = out; inw = 64;
        }
        // final output in bufB
    }

    // ---- enc_mean / enc_var: 4-layer bidir, H=64, finals -> mu / log_var --
    auto run_enc64 = [&](int basep, float* hTbase) {
        const float* in = bufB; int inw = 64;   // keep bufB (enc_rnn1 out) intact
        float* pp[2] = { bufA, bufC };
        for (int l = 0; l < 4; ++l) {
            const int b = basep + l * 8;
            gemm(in, inw, P(b + 0), P(b + 2), P(b + 3), gxa, BATCH * SEQ, 256, inw);
            gemm(in, inw, P(b + 4), P(b + 6), P(b + 7), gxb, BATCH * SEQ, 256, inw);
            float* out = pp[l & 1];
            lstm_rec<64><<<2, 256, 0, stream>>>(gxa, gxb, SEQ * 256, 256,
                                                P(b + 1), P(b + 5),
                                                out, 128, 0, 64,
                                                hTbase, 512, (l * 2 + 0) * 64, (l * 2 + 1) * 64,
                                                SEQ, 0, 1);
            in = out; inw = 128;
        }
    };
    run_enc64(ENCM, outMu);
    run_enc64(ENCV, outLv);

    // ---- z = mu + exp(0.5*log_var) * eps ---------------------------------
    reparam<<<(BATCH * 512 + 255) / 256, 256, 0, stream>>>(outMu, outLv, eps, zbuf, BATCH * 512);

    // ---- dec_rnn1: bidir H=32, input z constant over time ----------------
    gemm(zbuf, 512, P(DEC1 + 0), P(DEC1 + 2), P(DEC1 + 3), gxc0, BATCH, 128, 512);
    gemm(zbuf, 512, P(DEC1 + 4), P(DEC1 + 6), P(DEC1 + 7), gxc1, BATCH, 128, 512);
    lstm_rec<32><<<2, 256, 0, stream>>>(gxc0, gxc1, 128, 0,
                                        P(DEC1 + 1), P(DEC1 + 5),
                                        bufA, 64, 0, 32,
                                        nullptr, 0, 0, 0, SEQ, 0, 1);

    // ---- dec_rnn2: bidir H=64 --------------------------------------------
    gemm(bufA, 64, P(DEC2 + 0), P(DEC2 + 2), P(DEC2 + 3), gxa, BATCH * SEQ, 256, 64);
    gemm(bufA, 64, P(DEC2 + 4), P(DEC2 + 6), P(DEC2 + 7), gxb, BATCH * SEQ, 256, 64);
    lstm_rec<64><<<2, 256, 0, stream>>>(gxa, gxb, SEQ * 256, 256,
                                        P(DEC2 + 1), P(DEC2 + 5),
                                        bufB, 128, 0, 64,
                                        nullptr, 0, 0, 0, SEQ, 0, 1);

    // ---- dec_rnn3: unidir H=1 -> d ---------------------------------------
    lstm_h1<<<1, 128, 0, stream>>>(bufB, 128, P(DEC3 + 0), P(DEC3 + 1),
                                   P(DEC3 + 2), P(DEC3 + 3), outD, SEQ);
}